// SparseMeshUnpool_3719441678808
// MI455X (gfx1250) — compile-verified
//
#include <hip/hip_runtime.h>
#include <stdint.h>

// out = COO(rows, cols, vals) @ x   -- gather x[cols[i]], scale by vals[i],
// scatter-add into out[rows[i]].  Memory-bound (~0.5 FLOP/byte): coalesced
// 1KB row accesses, hardware f32 atomics at device scope, async global->LDS
// staging of the index triples (ASYNCcnt path), L2-resident x (51MB < 192MB).

#define F_DIM 256
#define NNZ_PER_BLOCK 64

typedef __attribute__((address_space(1))) int gint;   // global int
typedef __attribute__((address_space(3))) int lint;   // LDS int

#if __has_builtin(__builtin_amdgcn_global_load_async_to_lds_b32)
#define ASYNC_STAGE 1
#else
#define ASYNC_STAGE 0
#endif

__device__ __forceinline__ void wait_async_zero() {
#if __has_builtin(__builtin_amdgcn_s_wait_asynccnt)
  __builtin_amdgcn_s_wait_asynccnt(0);
#else
  asm volatile("s_wait_asynccnt 0x0" ::: "memory");
#endif
}

__global__ __launch_bounds__(256) void zero_out_kernel(float* __restrict__ out, int n) {
  int n4 = n >> 2;
  float4* __restrict__ out4 = reinterpret_cast<float4*>(out);
  const float4 z = make_float4(0.f, 0.f, 0.f, 0.f);
  int stride = gridDim.x * blockDim.x;
  for (int i = blockIdx.x * blockDim.x + threadIdx.x; i < n4; i += stride)
    out4[i] = z;
  if (blockIdx.x == 0 && threadIdx.x < (n & 3))
    out[(n4 << 2) + threadIdx.x] = 0.f;
}

__global__ __launch_bounds__(256) void scatter_add_kernel(
    const float* __restrict__ x,
    const float* __restrict__ vals,
    const int* __restrict__ rows,
    const int* __restrict__ cols,
    float* __restrict__ out,
    int nnz) {
  __shared__ float s_vals[NNZ_PER_BLOCK];
  __shared__ int   s_rows[NNZ_PER_BLOCK];
  __shared__ int   s_cols[NNZ_PER_BLOCK];

  const int tid  = threadIdx.x;
  const int base = blockIdx.x * NNZ_PER_BLOCK;

  if (base + NNZ_PER_BLOCK <= nnz) {
#if ASYNC_STAGE
    // CDNA5 async copy: per-lane global->LDS, tracked by ASYNCcnt.
    // Only waves 0-1 (tid<64) issue; they drain their own ASYNCcnt before
    // the workgroup barrier publishes the LDS contents.
    if (tid < NNZ_PER_BLOCK) {
      const int i = base + tid;
      __builtin_amdgcn_global_load_async_to_lds_b32(
          (gint*)(void*)(vals + i), (lint*)&s_vals[tid], 0, 0);
      __builtin_amdgcn_global_load_async_to_lds_b32(
          (gint*)(void*)(rows + i), (lint*)&s_rows[tid], 0, 0);
      __builtin_amdgcn_global_load_async_to_lds_b32(
          (gint*)(void*)(cols + i), (lint*)&s_cols[tid], 0, 0);
      wait_async_zero();
    }
#else
    if (tid < NNZ_PER_BLOCK) {
      const int i = base + tid;
      s_vals[tid] = vals[i];
      s_rows[tid] = rows[i];
      s_cols[tid] = cols[i];
    }
#endif
  } else {
    // tail block (never hit for NNZ=600000): guarded scalar staging
    if (tid < NNZ_PER_BLOCK) {
      const int  i  = base + tid;
      const bool ok = i < nnz;
      s_vals[tid] = ok ? vals[i] : 0.0f;  // val==0 -> +0.0 atomics, harmless
      s_rows[tid] = ok ? rows[i] : 0;
      s_cols[tid] = ok ? cols[i] : 0;
    }
  }
  __syncthreads();

  // 4 nonzeros in flight per block: 64 lanes x float4 = 256 floats = one row.
  const int sub  = tid >> 6;  // 0..3  : which nonzero of the group of 4
  const int lane = tid & 63;  // 0..63 : float4 lane within the 256-wide row
  const float4* __restrict__ x4 = reinterpret_cast<const float4*>(x);

  for (int j = 0; j < NNZ_PER_BLOCK / 4; ++j) {
    const int   li = (j << 2) + sub;
    const float v  = s_vals[li];
    const int   r  = s_rows[li];
    const int   c  = s_cols[li];

    // Warm the next random gather (L2 hit -> L0) with global_prefetch_b8.
    if (j + 1 < NNZ_PER_BLOCK / 4) {
      const int cn = s_cols[((j + 1) << 2) + sub];
      __builtin_prefetch(x4 + (size_t)cn * (F_DIM / 4) + lane, 0, 3);
    }

    const float4 xv = x4[(size_t)c * (F_DIM / 4) + lane];
    float4 p;
    p.x = v * xv.x;
    p.y = v * xv.y;
    p.z = v * xv.z;
    p.w = v * xv.w;

    // Hardware f32 atomic scatter, device scope (rows collide across WGPs).
    // No-return -> STOREcnt; s_endpgm's implicit wait-idle drains them.
    unsigned long long addr =
        (unsigned long long)(uintptr_t)(out + (size_t)r * F_DIM + (lane << 2));
    asm volatile("global_atomic_add_f32 %0, %1, off scope:SCOPE_DEV"
                 :: "v"(addr), "v"(p.x) : "memory");
    asm volatile("global_atomic_add_f32 %0, %1, off offset:4 scope:SCOPE_DEV"
                 :: "v"(addr), "v"(p.y) : "memory");
    asm volatile("global_atomic_add_f32 %0, %1, off offset:8 scope:SCOPE_DEV"
                 :: "v"(addr), "v"(p.z) : "memory");
    asm volatile("global_atomic_add_f32 %0, %1, off offset:12 scope:SCOPE_DEV"
                 :: "v"(addr), "v"(p.w) : "memory");
  }
}

extern "C" void kernel_launch(void* const* d_in, const int* in_sizes, int n_in,
                              void* d_out, int out_size, void* d_ws, size_t ws_size,
                              hipStream_t stream) {
  const float* x    = (const float*)d_in[0];
  const float* vals = (const float*)d_in[1];
  const int*   rows = (const int*)d_in[2];
  const int*   cols = (const int*)d_in[3];
  float*       out  = (float*)d_out;
  const int    nnz  = in_sizes[1];

  // 1) zero-init the (poisoned) output: ~205MB of streaming b128 stores
  hipLaunchKernelGGL(zero_out_kernel, dim3(4096), dim3(256), 0, stream,
                     out, out_size);

  // 2) gather-scale-scatter: one block per 64 nonzeros
  const int blocks = (nnz + NNZ_PER_BLOCK - 1) / NNZ_PER_BLOCK;
  hipLaunchKernelGGL(scatter_add_kernel, dim3(blocks), dim3(256), 0, stream,
                     x, vals, rows, cols, out, nnz);
}